// NeighborRadiusSearchLayer_90357521973573
// MI455X (gfx1250) — compile-verified
//
#include <hip/hip_runtime.h>
#include <hip/hip_bf16.h>

// Radius neighbor search on gfx1250:
//   d2[m][n] = |q_m|^2 + |r_n|^2 - 2 q_m . r_n
// computed per 16x16 tile with V_WMMA_F32_16X16X4_F32:
//   A (16x4, rows = queries)      = (qx, qy, qz, 1)
//   B (4x16, cols = refs)         = (-2rx, -2ry, -2rz, |r|^2)
//   C[m][n] = |r_n|^2 - 2 q_m.r_n   (sort key; |q_m|^2 folded into threshold)
// Main loop unrolled 2 ref tiles deep so both global_load_b96's issue before
// the first wait, with a speculative global_prefetch_b8 ahead. In-radius hits
// (rare: ~0.13/tile) are detected with a per-lane 8-bit mask + single branch,
// appended to per-query LDS lists via ds_add_rtn, then a ties-exact top-K
// selection reproduces jax.lax.top_k ordering.

#define K_MAX   64
#define QT      16    // queries per block (one WMMA row tile)
#define CAP     256   // per-query candidate capacity (E[neighbors] ~ 26)
#define WAVES   4
#define RADIUS2 0.0025f

typedef __attribute__((ext_vector_type(2))) float v2f;
typedef __attribute__((ext_vector_type(8))) float v8f;

__global__ __launch_bounds__(128)
void NeighborRadiusSearch_wmma_kernel(const float* __restrict__ ref,
                                      const float* __restrict__ query,
                                      int* __restrict__ out_idx,
                                      int* __restrict__ out_cnt,
                                      int nref, int nquery)
{
    __shared__ float s_key[QT][CAP];
    __shared__ int   s_idx[QT][CAP];
    __shared__ int   s_cnt[QT];
    __shared__ float s_qsq[QT];

    const int tid  = threadIdx.x;
    const int wave = tid >> 5;
    const int lane = tid & 31;
    const int m    = lane & 15;   // col-of-C per lane (ref within tile)
    const int half = lane >> 4;   // lane half selects K pair / row offset
    const int qbase = blockIdx.x * QT;

    // --- per-block setup: |q|^2 and counters -------------------------------
    if (tid < QT) {
        int q = qbase + tid;
        int qc = (q < nquery) ? q : (nquery - 1);
        float qx = query[3 * qc + 0];
        float qy = query[3 * qc + 1];
        float qz = query[3 * qc + 2];
        s_qsq[tid] = qx * qx + qy * qy + qz * qz;
        s_cnt[tid] = 0;
    }
    __syncthreads();

    // --- A operand: 16 queries x K=4, rows (qx,qy,qz,1) --------------------
    // ISA layout (32-bit A 16x4): v0 = K0 (lanes 0-15) / K2 (lanes 16-31),
    //                             v1 = K1 / K3.
    v2f a;
    {
        int q  = qbase + m;
        int qc = (q < nquery) ? q : (nquery - 1);
        float qx = query[3 * qc + 0];
        float qy = query[3 * qc + 1];
        float qz = query[3 * qc + 2];
        a.x = half ? qz : qx;
        a.y = half ? 1.0f : qy;
    }

    // Per-C-row threshold: within  <=>  C <= r^2 - |q|^2
    float thr[8];
#pragma unroll
    for (int i = 0; i < 8; ++i)
        thr[i] = RADIUS2 - s_qsq[i + 8 * half];

    const int ntiles = (nref + 15) >> 4;
    const int step   = WAVES;

    // B operand for ref tile t: K=4 x 16 refs, cols (-2rx,-2ry,-2rz,|r|^2)
    auto loadB = [&](int t) -> v2f {
        int n  = (t << 4) + m;
        int nc = (n < nref) ? n : (nref - 1);
        // speculative prefetch two unrolled steps ahead (OOB-safe)
        __builtin_prefetch(ref + 3 * (n + 32 * WAVES), 0, 1);
        float rx = ref[3 * nc + 0];
        float ry = ref[3 * nc + 1];
        float rz = ref[3 * nc + 2];
        float rsq = rx * rx + ry * ry + rz * rz;
        if (n >= nref) rsq = 1e30f;     // poison tail cols: key > thr always
        v2f b;
        b.x = half ? (-2.0f * rz) : (-2.0f * rx);
        b.y = half ? rsq          : (-2.0f * ry);
        return b;
    };

    // C VGPR i, this lane: row = i + 8*half, col = m, refidx = t*16 + m.
    // Build per-lane hit mask first so the (rare) append path costs one branch.
    auto scanHits = [&](const v8f& c, int t) {
        unsigned hm = 0u;
#pragma unroll
        for (int i = 0; i < 8; ++i)
            hm |= (c[i] <= thr[i]) ? (1u << i) : 0u;
        if (hm) {
#pragma unroll
            for (int i = 0; i < 8; ++i) {
                if (hm & (1u << i)) {
                    int row  = i + 8 * half;
                    int slot = atomicAdd(&s_cnt[row], 1);   // ds_add_rtn_u32
                    if (slot < CAP) {
                        s_key[row][slot] = c[i];
                        s_idx[row][slot] = (t << 4) + m;
                    }
                }
            }
        }
    };

    // --- main loop: 2 ref tiles (2 WMMAs) per iteration --------------------
    int t = wave;
    for (; t + step < ntiles; t += 2 * step) {
        v2f b0 = loadB(t);
        v2f b1 = loadB(t + step);
        v8f c0 = {};
        c0 = __builtin_amdgcn_wmma_f32_16x16x4_f32(
                false, a, false, b0, (short)0, c0, false, false);
        v8f c1 = {};
        c1 = __builtin_amdgcn_wmma_f32_16x16x4_f32(
                false, a, false, b1, (short)0, c1, false, false);
        scanHits(c0, t);
        scanHits(c1, t + step);
    }
    for (; t < ntiles; t += step) {
        v2f b0 = loadB(t);
        v8f c0 = {};
        c0 = __builtin_amdgcn_wmma_f32_16x16x4_f32(
                false, a, false, b0, (short)0, c0, false, false);
        scanHits(c0, t);
    }
    __syncthreads();

    // --- selection: top-K_MAX ascending by (key, idx), top_k tie rules -----
    if (tid < QT) {
        int q = qbase + tid;
        if (q < nquery) {
            int total  = s_cnt[tid];
            int stored = (total < CAP)   ? total : CAP;
            int nout   = (total < K_MAX) ? total : K_MAX;

            float lastK = -__builtin_huge_valf();
            int   lastI = -1;
            for (int k = 0; k < nout; ++k) {
                float bK = __builtin_huge_valf();
                int   bI = 0x7fffffff;
                for (int j = 0; j < stored; ++j) {
                    float kj = s_key[tid][j];
                    int   ij = s_idx[tid][j];
                    // strictly after the previously emitted (key, idx) ...
                    bool gt = (kj > lastK) || (kj == lastK && ij > lastI);
                    // ... and lexicographically minimal among remaining
                    bool lt = (kj < bK) || (kj == bK && ij < bI);
                    if (gt && lt) { bK = kj; bI = ij; }
                }
                out_idx[(size_t)q * K_MAX + k] = bI;
                lastK = bK; lastI = bI;
            }
            for (int k = nout; k < K_MAX; ++k)
                out_idx[(size_t)q * K_MAX + k] = -1;
            out_cnt[q] = nout;
        }
    }
}

extern "C" void kernel_launch(void* const* d_in, const int* in_sizes, int n_in,
                              void* d_out, int out_size, void* d_ws, size_t ws_size,
                              hipStream_t stream) {
    const float* ref   = (const float*)d_in[0];   // [nref, 3]  float32
    const float* query = (const float*)d_in[1];   // [nquery,3] float32
    int nref   = in_sizes[0] / 3;
    int nquery = in_sizes[1] / 3;

    // Output: int32 idx [nquery, K_MAX] then int32 cnt [nquery], flat.
    int* out_idx = (int*)d_out;
    int* out_cnt = out_idx + (size_t)nquery * K_MAX;

    int blocks = (nquery + QT - 1) / QT;  // 1250 for nquery=20000
    NeighborRadiusSearch_wmma_kernel<<<blocks, 128, 0, stream>>>(
        ref, query, out_idx, out_cnt, nref, nquery);
}